// Net_214748364926
// MI455X (gfx1250) — compile-verified
//
#include <hip/hip_runtime.h>

// GNN forward on MI455X (gfx1250, wave32).
// Memory/atomic bound. W1/W2 are pushed through segment_sum (per-row scale and
// right-multiply commute with scatter-add) to minimize per-edge traffic.
// Dense GEMMs use V_WMMA_F32_16X16X4_F32 (full-f32 WMMA, one wave = 16-row tile).
// Weights are pre-padded/transposed so B-fragment loads are unconditional
// contiguous b64 loads; tile epilogues have a wave-uniform full-tile fast path
// so no per-row exec divergence occurs on the common path.

typedef float v2f __attribute__((ext_vector_type(2)));
typedef float v8f __attribute__((ext_vector_type(8)));

// ---------------------------------------------------------------- utilities

__global__ void k_zero(float* __restrict__ p, long n) {
  long i = (long)blockIdx.x * blockDim.x + threadIdx.x;
  if (i < n) p[i] = 0.0f;
}

// Pad + transpose weights into column-major zero-padded layouts:
//   Wep[n*16 + k] = We[k*8  + n]  (n<8  else 0), 16x16
//   Wnp[n*64 + k] = Wn[k*16 + n]               , 16x64
//   W1p[n*24 + k] = W1[k*10 + n]  (n<10 else 0), 16x24
__global__ void k_prep_weights(const float* __restrict__ We,
                               const float* __restrict__ Wn,
                               const float* __restrict__ W1,
                               float* __restrict__ Wep, float* __restrict__ Wnp,
                               float* __restrict__ W1p) {
  int i = blockIdx.x * blockDim.x + threadIdx.x;
  if (i < 256) {
    int n = i >> 4, k = i & 15;
    Wep[i] = (n < 8) ? We[k * 8 + n] : 0.0f;
  }
  if (i < 1024) {
    int n = i >> 6, k = i & 63;
    Wnp[i] = Wn[k * 16 + n];
  }
  if (i < 384) {
    int n = i / 24, k = i % 24;
    W1p[i] = (n < 10) ? W1[k * 10 + n] : 0.0f;
  }
}

// deg_out[src]++, deg_in[dst]++ (degrees fit exactly in f32)
__global__ void k_degree(const int* __restrict__ src, const int* __restrict__ dst,
                         float* __restrict__ deg_out, float* __restrict__ deg_in, int E) {
  int e = blockIdx.x * blockDim.x + threadIdx.x;
  if (e < E) {
    atomicAdd(&deg_out[src[e]], 1.0f);
    atomicAdd(&deg_in[dst[e]], 1.0f);
  }
}

// ------------------------------------------------- edge GEMM + scatter (WMMA)
// 16-edge tile per wave: C(16x16) = ef(16x16) @ Wep(16x16, cols 8..15 zero),
// then atomic scatter row -> e_sum[dst[row]*8 + col].
__global__ void k_edge_gemm_scatter(const float* __restrict__ ef,
                                    const float* __restrict__ Wep,
                                    const int* __restrict__ dst,
                                    float* __restrict__ e_sum, int E) {
  int wave = blockIdx.x * (blockDim.x >> 5) + (threadIdx.x >> 5);
  int lane = threadIdx.x & 31;
  int e0 = wave * 16;
  if (e0 >= E) return;  // wave-uniform
  int m = lane & 15, kh = lane >> 4, col = m;
  int erow = e0 + m;
  int er = erow < E ? erow : (E - 1);      // clamped: address always valid
  float amask = (erow < E) ? 1.0f : 0.0f;  // zero OOB rows branchlessly

  v8f c = {};
#pragma unroll
  for (int kb = 0; kb < 16; kb += 4) {
    int k = kb + 2 * kh;
    v2f a = *reinterpret_cast<const v2f*>(ef + (size_t)er * 16 + k);
    a.x *= amask;
    a.y *= amask;
    v2f b = *reinterpret_cast<const v2f*>(Wep + col * 16 + k);
    c = __builtin_amdgcn_wmma_f32_16x16x4_f32(false, a, false, b, (short)0, c,
                                              false, false);
  }
  // C layout: VGPR v, lanes 0-15 -> M=v, lanes 16-31 -> M=v+8; N = lane&15.
  if (col < 8) {
    if (e0 + 16 <= E) {  // wave-uniform full-tile fast path: no per-row guards
      int d[8];
#pragma unroll
      for (int v = 0; v < 8; ++v) d[v] = dst[e0 + v + 8 * kh];
#pragma unroll
      for (int v = 0; v < 8; ++v)
        atomicAdd(&e_sum[(size_t)d[v] * 8 + col], c[v]);
    } else {
#pragma unroll
      for (int v = 0; v < 8; ++v) {
        int e = e0 + v + 8 * kh;
        if (e < E) atomicAdd(&e_sum[(size_t)dst[e] * 8 + col], c[v]);
      }
    }
  }
}

// ------------------------------------- per-node: norm coeffs + edge_h (cols 0-7)
// edge mean includes +be only when deg_in>0 (sum over edges carries be*deg_in).
__global__ void k_coeff_edgeh(const float* __restrict__ deg_out,
                              const float* __restrict__ deg_in,
                              const float* __restrict__ be,
                              const float* __restrict__ e_sum,
                              float* __restrict__ c_src, float* __restrict__ c_dst,
                              float* __restrict__ h24, int N) {
  int n = blockIdx.x * blockDim.x + threadIdx.x;
  if (n >= N) return;
  float dout = deg_out[n], din = deg_in[n];
  c_src[n] = (dout > 0.0f) ? rsqrtf(dout) : 0.0f;
  c_dst[n] = (din > 0.0f) ? rsqrtf(din) : 0.0f;
  float inv = 1.0f / fmaxf(din, 1.0f);
  float bsel = (din > 0.0f) ? 1.0f : 0.0f;
#pragma unroll
  for (int j = 0; j < 8; ++j) {
    float v = e_sum[(size_t)n * 8 + j] * inv + bsel * be[j];
    h24[(size_t)n * 24 + j] = fmaxf(v, 0.0f);
  }
}

// --------------------------------------- node GEMM (WMMA): relu(nf@Wn + bn)
// [N,64]@[64,16] -> h24 cols 8..23. 16 K-steps of 4.
__global__ void k_node_gemm(const float* __restrict__ nf,
                            const float* __restrict__ Wnp,
                            const float* __restrict__ bn,
                            float* __restrict__ h24, int N) {
  int wave = blockIdx.x * (blockDim.x >> 5) + (threadIdx.x >> 5);
  int lane = threadIdx.x & 31;
  int n0 = wave * 16;
  if (n0 >= N) return;
  int m = lane & 15, kh = lane >> 4, col = m;
  int nrow = n0 + m;
  int nr = nrow < N ? nrow : (N - 1);
  float amask = (nrow < N) ? 1.0f : 0.0f;

  v8f c = {};
#pragma unroll
  for (int kb = 0; kb < 64; kb += 4) {
    int k = kb + 2 * kh;
    v2f a = *reinterpret_cast<const v2f*>(nf + (size_t)nr * 64 + k);
    a.x *= amask;
    a.y *= amask;
    v2f b = *reinterpret_cast<const v2f*>(Wnp + col * 64 + k);
    c = __builtin_amdgcn_wmma_f32_16x16x4_f32(false, a, false, b, (short)0, c,
                                              false, false);
  }
  float bc = bn[col];
  if (n0 + 16 <= N) {  // wave-uniform fast path
#pragma unroll
    for (int v = 0; v < 8; ++v)
      h24[(size_t)(n0 + v + 8 * kh) * 24 + 8 + col] = fmaxf(c[v] + bc, 0.0f);
  } else {
#pragma unroll
    for (int v = 0; v < 8; ++v) {
      int nn = n0 + v + 8 * kh;
      if (nn < N) h24[(size_t)nn * 24 + 8 + col] = fmaxf(c[v] + bc, 0.0f);
    }
  }
}

// ------------------- conv1 pre-aggregation GEMM (WMMA): g1 = (h24*c_src)@W1
// [N,24]@[24,10] (cols 10..15 zero); c_src folded into A loads.
__global__ void k_g1_gemm(const float* __restrict__ h24,
                          const float* __restrict__ c_src,
                          const float* __restrict__ W1p,
                          float* __restrict__ g1, int N) {
  int wave = blockIdx.x * (blockDim.x >> 5) + (threadIdx.x >> 5);
  int lane = threadIdx.x & 31;
  int n0 = wave * 16;
  if (n0 >= N) return;
  int m = lane & 15, kh = lane >> 4, col = m;
  int nrow = n0 + m;
  int nr = nrow < N ? nrow : (N - 1);
  float cs = (nrow < N) ? c_src[nr] : 0.0f;  // doubles as OOB row mask

  v8f c = {};
#pragma unroll
  for (int kb = 0; kb < 24; kb += 4) {
    int k = kb + 2 * kh;
    v2f a = *reinterpret_cast<const v2f*>(h24 + (size_t)nr * 24 + k);
    a.x *= cs;
    a.y *= cs;
    v2f b = *reinterpret_cast<const v2f*>(W1p + col * 24 + k);
    c = __builtin_amdgcn_wmma_f32_16x16x4_f32(false, a, false, b, (short)0, c,
                                              false, false);
  }
  if (col < 10) {
    if (n0 + 16 <= N) {  // wave-uniform fast path
#pragma unroll
      for (int v = 0; v < 8; ++v)
        g1[(size_t)(n0 + v + 8 * kh) * 10 + col] = c[v];
    } else {
#pragma unroll
      for (int v = 0; v < 8; ++v) {
        int nn = n0 + v + 8 * kh;
        if (nn < N) g1[(size_t)nn * 10 + col] = c[v];
      }
    }
  }
}

// ------------------------------------------------ edge scatter-adds (10 / 2)
__global__ void k_scatter10(const int* __restrict__ src, const int* __restrict__ dst,
                            const float* __restrict__ g1, float* __restrict__ agg1, int E) {
  int e = blockIdx.x * blockDim.x + threadIdx.x;
  if (e >= E) return;
  size_t s = (size_t)src[e], d = (size_t)dst[e];
  const v2f* gs = reinterpret_cast<const v2f*>(g1 + s * 10);  // rows are 8B-aligned
  float* ad = agg1 + d * 10;
  v2f v0 = gs[0], v1 = gs[1], v2 = gs[2], v3 = gs[3], v4 = gs[4];
  atomicAdd(&ad[0], v0.x);
  atomicAdd(&ad[1], v0.y);
  atomicAdd(&ad[2], v1.x);
  atomicAdd(&ad[3], v1.y);
  atomicAdd(&ad[4], v2.x);
  atomicAdd(&ad[5], v2.y);
  atomicAdd(&ad[6], v3.x);
  atomicAdd(&ad[7], v3.y);
  atomicAdd(&ad[8], v4.x);
  atomicAdd(&ad[9], v4.y);
}

__global__ void k_scatter2(const int* __restrict__ src, const int* __restrict__ dst,
                           const float* __restrict__ g2, float* __restrict__ agg2, int E) {
  int e = blockIdx.x * blockDim.x + threadIdx.x;
  if (e >= E) return;
  size_t s = (size_t)src[e], d = (size_t)dst[e];
  v2f v = *reinterpret_cast<const v2f*>(g2 + s * 2);
  atomicAdd(&agg2[d * 2 + 0], v.x);
  atomicAdd(&agg2[d * 2 + 1], v.y);
}

// --------------- h1 = relu(agg1*c_dst + b1); g2 = (h1*c_src)@W2  (K=10 tiny)
__global__ void k_node2(const float* __restrict__ agg1,
                        const float* __restrict__ c_src, const float* __restrict__ c_dst,
                        const float* __restrict__ b1, const float* __restrict__ W2,
                        float* __restrict__ g2, int N) {
  int n = blockIdx.x * blockDim.x + threadIdx.x;
  if (n >= N) return;
  float cd = c_dst[n], cs = c_src[n];
  float a0 = 0.0f, a1 = 0.0f;
#pragma unroll
  for (int j = 0; j < 10; ++j) {
    float h = fmaxf(agg1[(size_t)n * 10 + j] * cd + b1[j], 0.0f);
    a0 += h * W2[j * 2 + 0];
    a1 += h * W2[j * 2 + 1];
  }
  g2[(size_t)n * 2 + 0] = a0 * cs;
  g2[(size_t)n * 2 + 1] = a1 * cs;
}

// ------------------------------------------------ out = agg2*c_dst + b2
__global__ void k_out(const float* __restrict__ agg2, const float* __restrict__ c_dst,
                      const float* __restrict__ b2, float* __restrict__ out, int N) {
  int n = blockIdx.x * blockDim.x + threadIdx.x;
  if (n >= N) return;
  float cd = c_dst[n];
  out[(size_t)n * 2 + 0] = agg2[(size_t)n * 2 + 0] * cd + b2[0];
  out[(size_t)n * 2 + 1] = agg2[(size_t)n * 2 + 1] * cd + b2[1];
}

// ---------------------------------------------------------------- launcher

extern "C" void kernel_launch(void* const* d_in, const int* in_sizes, int n_in,
                              void* d_out, int out_size, void* d_ws, size_t ws_size,
                              hipStream_t stream) {
  const float* node_feats = (const float*)d_in[0];
  const float* edge_feats = (const float*)d_in[1];
  const float* We = (const float*)d_in[2];
  const float* be = (const float*)d_in[3];
  const float* Wn = (const float*)d_in[4];
  const float* bn = (const float*)d_in[5];
  const float* W1 = (const float*)d_in[6];
  const float* b1 = (const float*)d_in[7];
  const float* W2 = (const float*)d_in[8];
  const float* b2 = (const float*)d_in[9];
  const int* src = (const int*)d_in[10];
  const int* dst = (const int*)d_in[11];
  float* out = (float*)d_out;

  const int N = in_sizes[0] / 64;
  const int E = in_sizes[10];
  const size_t Nl = (size_t)N;

  // Workspace layout (floats). Atomic targets first for one contiguous zero-fill.
  float* ws = (float*)d_ws;
  float* deg_in = ws;                  //  N
  float* deg_out = ws + Nl;            //  N
  float* e_sum = ws + 2 * Nl;          //  8N
  float* agg1 = ws + 10 * Nl;          // 10N
  float* agg2 = ws + 20 * Nl;          //  2N   -> zero region = 22N floats
  float* c_src = ws + 22 * Nl;         //  N
  float* c_dst = ws + 23 * Nl;         //  N
  float* h24 = ws + 24 * Nl;           // 24N
  float* g1 = ws + 48 * Nl;            // 10N
  float* g2 = ws + 58 * Nl;            //  2N
  float* Wep = ws + 60 * Nl;           //  256  (16x16 col-major, padded)
  float* Wnp = Wep + 256;              // 1024  (16x64 col-major)
  float* W1p = Wnp + 1024;             //  384  (16x24 col-major, padded)

  const int TB = 256;
  const long zn = 22L * N;
  k_zero<<<(int)((zn + TB - 1) / TB), TB, 0, stream>>>(ws, zn);
  k_prep_weights<<<4, TB, 0, stream>>>(We, Wn, W1, Wep, Wnp, W1p);

  const int eb = (E + TB - 1) / TB;
  const int nb = (N + TB - 1) / TB;
  const int waves_per_block = TB / 32;
  const int etile_blocks = (((E + 15) / 16) + waves_per_block - 1) / waves_per_block;
  const int ntile_blocks = (((N + 15) / 16) + waves_per_block - 1) / waves_per_block;

  k_degree<<<eb, TB, 0, stream>>>(src, dst, deg_out, deg_in, E);
  k_edge_gemm_scatter<<<etile_blocks, TB, 0, stream>>>(edge_feats, Wep, dst, e_sum, E);
  k_coeff_edgeh<<<nb, TB, 0, stream>>>(deg_out, deg_in, be, e_sum, c_src, c_dst, h24, N);
  k_node_gemm<<<ntile_blocks, TB, 0, stream>>>(node_feats, Wnp, bn, h24, N);
  k_g1_gemm<<<ntile_blocks, TB, 0, stream>>>(h24, c_src, W1p, g1, N);
  k_scatter10<<<eb, TB, 0, stream>>>(src, dst, g1, agg1, E);
  k_node2<<<nb, TB, 0, stream>>>(agg1, c_src, c_dst, b1, W2, g2, N);
  k_scatter2<<<eb, TB, 0, stream>>>(src, dst, g2, agg2, E);
  k_out<<<nb, TB, 0, stream>>>(agg2, c_dst, b2, out, N);
}